// HANGraphClassifier_82033875353652
// MI455X (gfx1250) — compile-verified
//
#include <hip/hip_runtime.h>
#include <hip/hip_bf16.h>
#include <math.h>

typedef float v2f __attribute__((ext_vector_type(2)));
typedef float v8f __attribute__((ext_vector_type(8)));

#define N_PROC_C 100000
#define HID 64
#define BSZ 128

// workspace layout (bytes), all offsets 64B-aligned
#define OFF_GATE_FP 0        // proc gates from file->proc edges (100096)
#define OFF_GATE_SP 100096   // proc gates from sock->proc edges (100096)
#define OFF_GATE_PF 200192   // file gates (100096)
#define OFF_GATE_PS 300288   // sock gates (50048)
#define OFF_CNT     350336   // 7 * 128 ints = 3584
#define OFF_G       353920   // 128*64 floats = 32768
#define CLEAR_WORDS (353920 / 4)

// ---------------- clear workspace (gates + counts) ----------------
__global__ void k_clear(unsigned int* w, int nwords) {
    int i = blockIdx.x * blockDim.x + threadIdx.x;
    int stride = gridDim.x * blockDim.x;
    for (; i < nwords; i += stride) w[i] = 0u;
}

// ---------------- mark dst gates: gate[dst[i]] = 1 ----------------
__global__ void k_mark(const int* __restrict__ dst, int n, unsigned char* __restrict__ gate) {
    int tid = blockIdx.x * blockDim.x + threadIdx.x;
    int stride = gridDim.x * blockDim.x;
    int n4 = n >> 2;
    const int4* d4 = (const int4*)dst;
    for (int i = tid; i < n4; i += stride) {
        __builtin_prefetch(d4 + i + stride, 0, 0);  // global_prefetch_b8
        int4 v = d4[i];
        gate[v.x] = 1; gate[v.y] = 1; gate[v.z] = 1; gate[v.w] = 1;
    }
    for (int i = (n4 << 2) + tid; i < n; i += stride) gate[dst[i]] = 1;
}

// ------- per-batch counts: total, gated-A, (optional) gated-B -------
__global__ void k_count(const unsigned char* __restrict__ gA,
                        const unsigned char* __restrict__ gB,
                        const int* __restrict__ batch, int n,
                        int* cA, int* cB, int* cT) {
    __shared__ int h[3 * BSZ];
    for (int t = threadIdx.x; t < 3 * BSZ; t += blockDim.x) h[t] = 0;
    __syncthreads();
    int stride = gridDim.x * blockDim.x;
    for (int i = blockIdx.x * blockDim.x + threadIdx.x; i < n; i += stride) {
        int b = batch[i];
        atomicAdd(&h[b], 1);
        if (gA[i]) atomicAdd(&h[BSZ + b], 1);
        if (gB != nullptr && gB[i]) atomicAdd(&h[2 * BSZ + b], 1);
    }
    __syncthreads();
    if (threadIdx.x < BSZ) {
        int b = threadIdx.x;
        if (h[b])           atomicAdd(&cT[b], h[b]);
        if (h[BSZ + b])     atomicAdd(&cA[b], h[BSZ + b]);
        if (gB != nullptr && h[2 * BSZ + b]) atomicAdd(&cB[b], h[2 * BSZ + b]);
    }
}

__device__ float block_reduce128(float* red, int tid, float v) {
    red[tid] = v;
    __syncthreads();
    for (int s = 64; s > 0; s >>= 1) {
        if (tid < s) red[tid] += red[tid + s];
        __syncthreads();
    }
    float r = red[0];
    __syncthreads();
    return r;
}

// ---- projections, semantic attention softmax, build g[128][64] ----
__global__ void k_prep(const float* __restrict__ node_emb, const float* __restrict__ proj_w,
                       const float* __restrict__ proj_b, const float* __restrict__ k_w,
                       const float* __restrict__ k_b, const float* __restrict__ q_vec,
                       const int* __restrict__ cnt, float* __restrict__ g_out) {
    __shared__ float r[3][HID];     // r_p, r_f, r_s  (relu'd projected type vectors)
    __shared__ float red[BSZ];
    __shared__ float sc[2];
    int tid = threadIdx.x;

    if (tid < HID) {
        for (int t = 0; t < 3; ++t) {
            float acc = proj_b[t * HID + tid];
            for (int j = 0; j < HID; ++j)
                acc += node_emb[t * HID + j] * proj_w[(t * HID + tid) * HID + j];
            r[t][tid] = fmaxf(acc, 0.f);
        }
    }
    __syncthreads();

    float c1 = block_reduce128(red, tid, (float)cnt[0 * BSZ + tid]);  // #proc gated by f->p
    float c2 = block_reduce128(red, tid, (float)cnt[1 * BSZ + tid]);  // #proc gated by s->p

    // semantic scores over [o_proc1, o_proc2]
    float contrib1 = 0.f, contrib2 = 0.f;
    if (tid < HID) {
        float af = k_b[tid], as_ = k_b[tid];
        for (int j = 0; j < HID; ++j) {
            af  += r[1][j] * k_w[tid * HID + j];
            as_ += r[2][j] * k_w[tid * HID + j];
        }
        float tf = tanhf(af), ts = tanhf(as_), t0 = tanhf(k_b[tid]);
        const float invN = 1.0f / (float)N_PROC_C;
        contrib1 = q_vec[tid] * ((c1 * tf + ((float)N_PROC_C - c1) * t0) * invN);
        contrib2 = q_vec[tid] * ((c2 * ts + ((float)N_PROC_C - c2) * t0) * invN);
    }
    float s1 = block_reduce128(red, tid, contrib1);
    float s2 = block_reduce128(red, tid, contrib2);
    if (tid == 0) {
        float mx = fmaxf(s1, s2);
        float e1 = expf(s1 - mx), e2 = expf(s2 - mx);
        float inv = 1.0f / (e1 + e2);
        sc[0] = e1 * inv; sc[1] = e2 * inv;
    }
    __syncthreads();
    float a1 = sc[0], a2 = sc[1];

    // g[b] = (pool_proc + pool_file + pool_sock) / 3
    int b = tid;
    float c1b = (float)cnt[0 * BSZ + b];
    float c2b = (float)cnt[1 * BSZ + b];
    float pt  = fmaxf((float)cnt[2 * BSZ + b], 1.f);
    float fg  = (float)cnt[3 * BSZ + b];
    float ft  = fmaxf((float)cnt[4 * BSZ + b], 1.f);
    float sg_ = (float)cnt[5 * BSZ + b];
    float st  = fmaxf((float)cnt[6 * BSZ + b], 1.f);
    for (int f = 0; f < HID; ++f) {
        float v = (c1b * a1 * r[1][f] + c2b * a2 * r[2][f]) / pt
                + (fg  * r[0][f]) / ft
                + (sg_ * r[0][f]) / st;
        g_out[b * HID + f] = v * (1.0f / 3.0f);
    }
}

// ---- classifier MLP: relu(g@W1^T + b1)@W2^T + b2 via V_WMMA_F32_16X16X4_F32 ----
__global__ void __launch_bounds__(256)
k_classifier(const float* __restrict__ g, const float* __restrict__ w1,
             const float* __restrict__ b1, const float* __restrict__ w2,
             const float* __restrict__ b2, float* __restrict__ out) {
    __shared__ float sg[128 * 64];   // g, later reused for hidden
    __shared__ float sw1[64 * 64];
    int tid = threadIdx.x;
    for (int i = tid; i < 128 * 64; i += 256) sg[i] = g[i];
    for (int i = tid; i < 64 * 64;  i += 256) sw1[i] = w1[i];
    __syncthreads();

    int wave = tid >> 5, lane = tid & 31;
    int lm = lane & 15;              // M (for A) / N (for B) within tile
    int kh = (lane >> 4) << 1;       // lane half selects K offset 0 / 2
    int m0 = wave << 4;              // 8 waves * 16 rows = 128 rows

    v8f acc[4] = {{}, {}, {}, {}};
#pragma unroll
    for (int k0 = 0; k0 < 64; k0 += 4) {
        v2f a;
        a.x = sg[(m0 + lm) * 64 + k0 + kh];
        a.y = sg[(m0 + lm) * 64 + k0 + kh + 1];
#pragma unroll
        for (int nt = 0; nt < 4; ++nt) {
            v2f bb;
            int n = nt * 16 + lm;
            bb.x = sw1[n * 64 + k0 + kh];      // B[k][n] = w1[n][k]
            bb.y = sw1[n * 64 + k0 + kh + 1];
            acc[nt] = __builtin_amdgcn_wmma_f32_16x16x4_f32(
                false, a, false, bb, (short)0, acc[nt], false, false);
        }
    }
    __syncthreads();
    // hidden = relu(acc + b1) scattered back into sg (C/D layout: VGPR r -> M=r(+8))
#pragma unroll
    for (int nt = 0; nt < 4; ++nt) {
        int n = nt * 16 + lm;
        float bias = b1[n];
#pragma unroll
        for (int rr = 0; rr < 8; ++rr) {
            int m = m0 + rr + ((lane >> 4) << 3);
            sg[m * 64 + n] = fmaxf(acc[nt][rr] + bias, 0.f);
        }
    }
    __syncthreads();
    // out[128,2] = hidden @ w2^T + b2
    int m = tid >> 1, c = tid & 1;
    float s = b2[c];
    for (int k = 0; k < 64; ++k) s += sg[m * 64 + k] * w2[c * 64 + k];
    out[m * 2 + c] = s;
}

extern "C" void kernel_launch(void* const* d_in, const int* in_sizes, int n_in,
                              void* d_out, int out_size, void* d_ws, size_t ws_size,
                              hipStream_t stream) {
    const float* node_emb = (const float*)d_in[0];
    const float* proj_w   = (const float*)d_in[1];
    const float* proj_b   = (const float*)d_in[2];
    // d_in[3]/d_in[4] (a_src/a_dst) cancel exactly: per-type-constant features
    // make every edge logit identical, so the edge softmax is uniform.
    const float* k_w   = (const float*)d_in[5];
    const float* k_b   = (const float*)d_in[6];
    const float* q_vec = (const float*)d_in[7];
    const float* w1 = (const float*)d_in[8];
    const float* b1 = (const float*)d_in[9];
    const float* w2 = (const float*)d_in[10];
    const float* b2 = (const float*)d_in[11];
    const int* pf_dst = (const int*)d_in[13];  // src arrays unused by the math
    const int* fp_dst = (const int*)d_in[15];
    const int* ps_dst = (const int*)d_in[17];
    const int* sp_dst = (const int*)d_in[19];
    const int* batch_p = (const int*)d_in[20];
    const int* batch_f = (const int*)d_in[21];
    const int* batch_s = (const int*)d_in[22];
    int e_pf = in_sizes[13], e_fp = in_sizes[15], e_ps = in_sizes[17], e_sp = in_sizes[19];
    int np = in_sizes[20], nf = in_sizes[21], ns = in_sizes[22];

    char* ws = (char*)d_ws;
    unsigned char* gate_fp = (unsigned char*)(ws + OFF_GATE_FP);
    unsigned char* gate_sp = (unsigned char*)(ws + OFF_GATE_SP);
    unsigned char* gate_pf = (unsigned char*)(ws + OFF_GATE_PF);
    unsigned char* gate_ps = (unsigned char*)(ws + OFF_GATE_PS);
    int*   cnt = (int*)(ws + OFF_CNT);
    float* g   = (float*)(ws + OFF_G);
    float* out = (float*)d_out;

    k_clear<<<(CLEAR_WORDS + 255) / 256, 256, 0, stream>>>((unsigned int*)ws, CLEAR_WORDS);

    k_mark<<<1024, 256, 0, stream>>>(fp_dst, e_fp, gate_fp);
    k_mark<<<1024, 256, 0, stream>>>(sp_dst, e_sp, gate_sp);
    k_mark<<<1024, 256, 0, stream>>>(pf_dst, e_pf, gate_pf);
    k_mark<<<512,  256, 0, stream>>>(ps_dst, e_ps, gate_ps);

    // proc: gated by fp (cnt0) and sp (cnt1), total cnt2
    k_count<<<256, 256, 0, stream>>>(gate_fp, gate_sp, batch_p, np,
                                     cnt + 0 * BSZ, cnt + 1 * BSZ, cnt + 2 * BSZ);
    // file: gated by pf (cnt3), total cnt4
    k_count<<<256, 256, 0, stream>>>(gate_pf, (const unsigned char*)nullptr, batch_f, nf,
                                     cnt + 3 * BSZ, (int*)nullptr, cnt + 4 * BSZ);
    // sock: gated by ps (cnt5), total cnt6
    k_count<<<128, 256, 0, stream>>>(gate_ps, (const unsigned char*)nullptr, batch_s, ns,
                                     cnt + 5 * BSZ, (int*)nullptr, cnt + 6 * BSZ);

    k_prep<<<1, 128, 0, stream>>>(node_emb, proj_w, proj_b, k_w, k_b, q_vec, cnt, g);
    k_classifier<<<1, 256, 0, stream>>>(g, w1, b1, w2, b2, out);
}